// CGN_16827681865781
// MI455X (gfx1250) — compile-verified
//
#include <hip/hip_runtime.h>
#include <hip/hip_bf16.h>

typedef __attribute__((ext_vector_type(2))) float v2f;
typedef __attribute__((ext_vector_type(4))) float v4f;
typedef __attribute__((ext_vector_type(8))) float v8f;

namespace {
constexpr int kBatch  = 4096;
constexpr int kU      = 20;          // DIM_U1 == DIM_U2
constexpr int kDZ     = 6;
constexpr int kNZ     = kDZ * kU;    // 120
constexpr int kMTotal = kBatch * kU; // 81920 fused (batch, u) rows
constexpr int kWaves  = 8;           // waves per block (256 threads)

// Output regions (flat float offsets, tuple order f1, g1, f2, g2)
constexpr int F1_OFF = 0;
constexpr int G1_OFF = kMTotal;                   // 81,920
constexpr int F2_OFF = G1_OFF + kMTotal * kNZ;    // 9,912,320
constexpr int G2_OFF = F2_OFF + kBatch * kNZ;     // 10,403,840
} // namespace

// ---------------------------------------------------------------------------
// fp32 WMMA: D(16x16) = A(16x4) * B(4x16) + C   (V_WMMA_F32_16X16X4_F32)
// ---------------------------------------------------------------------------
__device__ __forceinline__ v8f wmma4(v2f a, v2f b, v8f c) {
  return __builtin_amdgcn_wmma_f32_16x16x4_f32(
      /*neg_a=*/false, a, /*neg_b=*/false, b,
      /*c_mod=*/(short)0, c, /*reuse_a=*/false, /*reuse_b=*/false);
}

// One 16x16 output tile of act(16xK) @ W(KxN), K chained in steps of 4.
// A layout: lane = M row (lane&15); VGPR0/1 = K = 2h, 2h+1 (h = lane>>4).
// B layout: lane = N col (lane&15); VGPR0/1 = K = 2h, 2h+1.
template <int K>
__device__ __forceinline__ v8f gemm_tile(const float* __restrict__ actA, int sA,
                                         const float* __restrict__ W, int sW,
                                         int n0, int lane) {
  const int rn = lane & 15;
  const int h  = lane >> 4;
  v8f c = {};
#pragma unroll
  for (int k0 = 0; k0 < K; k0 += 4) {
    v2f a, b;
    a.x = actA[rn * sA + k0 + 2 * h];
    a.y = actA[rn * sA + k0 + 2 * h + 1];
    b.x = W[(k0 + 2 * h) * sW + n0 + rn];
    b.y = W[(k0 + 2 * h + 1) * sW + n0 + rn];
    c = wmma4(a, b, c);
  }
  return c;
}

// C/D layout: VGPR j, lane half h -> row = j + 8*h, col = lane&15.
__device__ __forceinline__ void store_act_relu(v8f d, float* __restrict__ out,
                                               int s, const float* __restrict__ bias,
                                               int n0, int lane) {
  const int n = lane & 15;
  const int h = lane >> 4;
  const float bv = bias[n0 + n];
#pragma unroll
  for (int j = 0; j < 8; ++j) {
    float v = d[j] + bv;
    out[(j + 8 * h) * s + n0 + n] = v > 0.f ? v : 0.f;
  }
}

// ---------------------------------------------------------------------------
// Pass 1: non-temporal wide zero-fill of the whole output (dominant traffic).
// ---------------------------------------------------------------------------
__global__ __launch_bounds__(256) void CGN_fill_zero(float* __restrict__ out, int n) {
  const int n4 = n >> 2;
  v4f z = {};
  v4f* p = (v4f*)out;
  for (int i = blockIdx.x * blockDim.x + threadIdx.x; i < n4;
       i += gridDim.x * blockDim.x)
    __builtin_nontemporal_store(z, p + i);          // global_store_b128 th:NT
  if (blockIdx.x == 0 && threadIdx.x == 0)
    for (int i = n & ~3; i < n; ++i) out[i] = 0.f;
}

// ---------------------------------------------------------------------------
// Pass 2: WMMA MLP chain + scatter of nonzeros. One 16-row M-tile per wave.
// ---------------------------------------------------------------------------
__global__ __launch_bounds__(256) void CGN_mlp_wmma(
    const float* __restrict__ x,
    const float* __restrict__ W1a, const float* __restrict__ b1a,
    const float* __restrict__ W1b, const float* __restrict__ b1b,
    const float* __restrict__ W1c, const float* __restrict__ b1c,
    const float* __restrict__ W1d, const float* __restrict__ b1d,
    const float* __restrict__ W2a, const float* __restrict__ b2a,
    const float* __restrict__ W2b, const float* __restrict__ b2b,
    const float* __restrict__ W2c, const float* __restrict__ b2c,
    const float* __restrict__ W2d, const float* __restrict__ b2d,
    float* __restrict__ out) {
  // Weights staged in LDS; K padded 3->4, N padded 19->32 / 186->192 with 0.
  __shared__ float s_w1a[4 * 16],  s_b1a[16];
  __shared__ float s_w1b[16 * 32], s_b1b[32];
  __shared__ float s_w1c[32 * 16], s_b1c[16];
  __shared__ float s_w1d[16 * 32], s_b1d[32];
  __shared__ float s_w2a[4 * 16],  s_b2a[16];
  __shared__ float s_w2b[16 * 32], s_b2b[32];
  __shared__ float s_w2c[32 * 16], s_b2c[16];
  __shared__ float s_w2d[16 * 192], s_b2d[192];
  // Per-wave activation staging (bank-conflict padded strides 17 / 33).
  __shared__ float s_acta[kWaves][16 * 17];
  __shared__ float s_actb[kWaves][16 * 33];

  const int tid = threadIdx.x;
  for (int i = tid; i < 64; i += 256) {
    int k = i >> 4, n = i & 15;
    s_w1a[i] = (k < 3) ? W1a[k * 16 + n] : 0.f;
    s_w2a[i] = (k < 3) ? W2a[k * 16 + n] : 0.f;
  }
  for (int i = tid; i < 512; i += 256) {
    s_w1b[i] = W1b[i]; s_w2b[i] = W2b[i];
    s_w1c[i] = W1c[i]; s_w2c[i] = W2c[i];
    int k = i >> 5, n = i & 31;
    s_w1d[i] = (n < 19) ? W1d[k * 19 + n] : 0.f;
  }
  for (int i = tid; i < 16 * 192; i += 256) {
    int k = i / 192, n = i % 192;
    s_w2d[i] = (n < 186) ? W2d[k * 186 + n] : 0.f;
  }
  for (int i = tid; i < 16; i += 256) {
    s_b1a[i] = b1a[i]; s_b2a[i] = b2a[i];
    s_b1c[i] = b1c[i]; s_b2c[i] = b2c[i];
  }
  for (int i = tid; i < 32; i += 256) {
    s_b1b[i] = b1b[i]; s_b2b[i] = b2b[i];
    s_b1d[i] = (i < 19) ? b1d[i] : 0.f;
  }
  for (int i = tid; i < 192; i += 256) s_b2d[i] = (i < 186) ? b2d[i] : 0.f;
  __syncthreads();

  const int lane = tid & 31;
  const int wave = tid >> 5;
  const int tile = blockIdx.x * kWaves + wave;
  const int m0 = tile * 16;
  if (m0 >= kMTotal) return;  // uniform per wave

  float* acta = s_acta[wave];
  float* actb = s_actb[wave];
  const int rn = lane & 15;
  const int h  = lane >> 4;

  // Layer-1 A fragment directly from global x: xl[r][k] = x[b, (u+k-1) mod 20],
  // k in {0,1,2}, K padded to 4 (k==3 -> 0).  a = {xl[2h], xl[2h+1]}.
  const int m  = m0 + rn;
  const int bm = m / kU;
  const int um = m % kU;
  v2f a1;
  a1.x = x[bm * kU + (um + 2 * h + kU - 1) % kU];
  a1.y = (h == 0) ? x[bm * kU + um] : 0.f;

  // ============================ MLP 1 =====================================
  {
    v2f bf;
    bf.x = s_w1a[(2 * h) * 16 + rn];
    bf.y = s_w1a[(2 * h + 1) * 16 + rn];
    v8f c0 = {};
    c0 = wmma4(a1, bf, c0);
    store_act_relu(c0, acta, 17, s_b1a, 0, lane);                 // h1: 16x16

    v8f h2a = gemm_tile<16>(acta, 17, s_w1b, 32, 0, lane);        // h2: 16x32
    v8f h2b = gemm_tile<16>(acta, 17, s_w1b, 32, 16, lane);
    store_act_relu(h2a, actb, 33, s_b1b, 0, lane);
    store_act_relu(h2b, actb, 33, s_b1b, 16, lane);

    v8f h3 = gemm_tile<32>(actb, 33, s_w1c, 16, 0, lane);         // h3: 16x16
    store_act_relu(h3, acta, 17, s_b1c, 0, lane);

#pragma unroll
    for (int nt = 0; nt < 2; ++nt) {                              // out1: 16x19
      const int n0 = nt * 16;
      v8f d = gemm_tile<16>(acta, 17, s_w1d, 32, n0, lane);
      const int cg = n0 + rn;
      if (cg < 19) {
        const float bv = s_b1d[cg];
#pragma unroll
        for (int j = 0; j < 8; ++j) {
          const int mm = m0 + j + 8 * h;
          const int uj = mm % kU;
          const float v = d[j] + bv;
          if (cg == 0) {
            out[F1_OFF + mm] = v;                                 // f1[b,u,0]
          } else {
            const int col = (cg - 1 + (uj - 1) * kDZ + kNZ) % kNZ;
            out[G1_OFF + mm * kNZ + col] = v;                     // g1 band
          }
        }
      }
    }
  }

  // ============================ MLP 2 =====================================
  {
    v2f bf;
    bf.x = s_w2a[(2 * h) * 16 + rn];
    bf.y = s_w2a[(2 * h + 1) * 16 + rn];
    v8f c0 = {};
    c0 = wmma4(a1, bf, c0);
    store_act_relu(c0, acta, 17, s_b2a, 0, lane);

    v8f h2a = gemm_tile<16>(acta, 17, s_w2b, 32, 0, lane);
    v8f h2b = gemm_tile<16>(acta, 17, s_w2b, 32, 16, lane);
    store_act_relu(h2a, actb, 33, s_b2b, 0, lane);
    store_act_relu(h2b, actb, 33, s_b2b, 16, lane);

    v8f h3 = gemm_tile<32>(actb, 33, s_w2c, 16, 0, lane);
    store_act_relu(h3, acta, 17, s_b2c, 0, lane);

#pragma unroll
    for (int nt = 0; nt < 12; ++nt) {                             // out2: 16x186
      const int n0 = nt * 16;
      v8f d = gemm_tile<16>(acta, 17, s_w2d, 192, n0, lane);
      const int cg = n0 + rn;
      if (cg < 186) {
        const float bv = s_b2d[cg];
#pragma unroll
        for (int j = 0; j < 8; ++j) {
          const int mm = m0 + j + 8 * h;
          const int bj = mm / kU, uj = mm % kU;
          const float v = d[j] + bv;
          if (cg < kDZ) {
            out[F2_OFF + bj * kNZ + uj * kDZ + cg] = v;           // f2
          } else {
            const int t = cg - kDZ;
            const int q = t / 30, c2 = t % 30;
            const int col = (c2 + (uj - 2) * kDZ + kNZ) % kNZ;
            out[G2_OFF + bj * (kNZ * kNZ) + (uj * kDZ + q) * kNZ + col] = v;
          }
        }
      }
    }
  }
}

// ---------------------------------------------------------------------------
extern "C" void kernel_launch(void* const* d_in, const int* in_sizes, int n_in,
                              void* d_out, int out_size, void* d_ws, size_t ws_size,
                              hipStream_t stream) {
  (void)in_sizes; (void)n_in; (void)d_ws; (void)ws_size;
  const float* x   = (const float*)d_in[0];
  const float* W1a = (const float*)d_in[1];  const float* b1a = (const float*)d_in[2];
  const float* W1b = (const float*)d_in[3];  const float* b1b = (const float*)d_in[4];
  const float* W1c = (const float*)d_in[5];  const float* b1c = (const float*)d_in[6];
  const float* W1d = (const float*)d_in[7];  const float* b1d = (const float*)d_in[8];
  const float* W2a = (const float*)d_in[9];  const float* b2a = (const float*)d_in[10];
  const float* W2b = (const float*)d_in[11]; const float* b2b = (const float*)d_in[12];
  const float* W2c = (const float*)d_in[13]; const float* b2c = (const float*)d_in[14];
  const float* W2d = (const float*)d_in[15]; const float* b2d = (const float*)d_in[16];
  float* out = (float*)d_out;

  // Pass 1: zero-fill 277 MB of output with NT b128 stores (bandwidth bound).
  CGN_fill_zero<<<2048, 256, 0, stream>>>(out, out_size);

  // Pass 2: 5120 M-tiles, 8 waves/block -> 640 blocks of 256 threads.
  CGN_mlp_wmma<<<kMTotal / 16 / kWaves, 32 * kWaves, 0, stream>>>(
      x, W1a, b1a, W1b, b1b, W1c, b1c, W1d, b1d,
      W2a, b2a, W2b, b2b, W2c, b2c, W2d, b2d, out);
}